// VNNGP_75153337745806
// MI455X (gfx1250) — compile-verified
//
#include <hip/hip_runtime.h>

#define N_PTS  131072
#define M_IND  512
#define DIM    8
#define K_NN   16
#define JITTER 0.0001f

typedef __attribute__((ext_vector_type(2))) float v2f;
typedef __attribute__((ext_vector_type(8))) float v8f;

// D = A(16x4, f32) * B(4x16, f32) + C(16x16, f32) : exact-precision CDNA5 matrix path
__device__ __forceinline__ v8f wmma_f32_k4(v2f a, v2f b, v8f c) {
  return __builtin_amdgcn_wmma_f32_16x16x4_f32(false, a, false, b, (short)0, c, false, false);
}

__device__ __forceinline__ v2f ld2(const float* __restrict__ p) {
  return *(const v2f*)p;   // 8B-aligned at all call sites -> global_load_b64
}

// ---------------------------------------------------------------------------
// Kernel 0: materialize Lu_t = tril(Lu,-1)+diag(exp(diag(Lu))) densely (so the
// GEMM fragment loads are branch-free b64 loads), and precompute zsq[512].
// ---------------------------------------------------------------------------
__global__ void __launch_bounds__(256)
prep_kernel(const float* __restrict__ Lu, const float* __restrict__ Z,
            float* __restrict__ Lut, float* __restrict__ zsq) {
  const int e = blockIdx.x * 256 + threadIdx.x;     // 0 .. 512*512-1
  const int r = e >> 9, c = e & (M_IND - 1);
  float v = 0.f;
  if (c < r)       v = Lu[e];
  else if (c == r) v = __expf(Lu[e]);
  Lut[e] = v;
  if (e < M_IND) {
    float s = 0.f;
    #pragma unroll
    for (int d = 0; d < DIM; ++d) s = fmaf(Z[e * DIM + d], Z[e * DIM + d], s);
    zsq[e] = s;
  }
}

// ---------------------------------------------------------------------------
// Kernel 1: Kzz_j = var*exp(-0.5*sqdist(Z,Z)/ls2) + jitter*I   (== L@L.T)
//           Dm    = Lu_t@Lu_t.T - Kzz_j                         (== S - Kzz)
// One wave per 16x16 tile. Inner loop = 2x global_load_b64 + 1 WMMA.
// ---------------------------------------------------------------------------
__global__ void __launch_bounds__(32)
build_mats_kernel(const float* __restrict__ Z, const float* __restrict__ Lut,
                  const float* __restrict__ zsq,
                  const float* __restrict__ pls, const float* __restrict__ pvar,
                  float* __restrict__ Kzz, float* __restrict__ Dm) {
  const int ti = blockIdx.x, tj = blockIdx.y;
  const int lane = threadIdx.x;
  const int half = lane >> 4, l16 = lane & 15;
  const float ls2 = pls[0] * pls[0];
  const float var = pvar[0];
  const int arow = ti * 16 + l16;   // A-fragment row (M = lane&15)
  const int brow = tj * 16 + l16;   // B-fragment col (N = lane&15)

  // S tile: Lu_t @ Lu_t^T over K=512 (128 chained v_wmma_f32_16x16x4_f32)
  const float* __restrict__ pa = Lut + arow * M_IND + half * 2;
  const float* __restrict__ pb = Lut + brow * M_IND + half * 2;
  v8f accS = {};
  #pragma unroll 4
  for (int k = 0; k < M_IND; k += 4) {
    accS = wmma_f32_k4(ld2(pa + k), ld2(pb + k), accS);
  }

  // Z @ Z^T tile over K=8 (two WMMAs)
  v8f dotz = {};
  dotz = wmma_f32_k4(ld2(Z + arow * DIM + half * 2),
                     ld2(Z + brow * DIM + half * 2), dotz);
  dotz = wmma_f32_k4(ld2(Z + arow * DIM + 4 + half * 2),
                     ld2(Z + brow * DIM + 4 + half * 2), dotz);

  // C/D layout: VGPR v -> M = v + 8*half, N = lane&15
  const int n = tj * 16 + l16;
  const float zn2 = zsq[n];
  #pragma unroll
  for (int v = 0; v < 8; ++v) {
    const int m = ti * 16 + v + half * 8;
    const float d2 = zsq[m] + zn2 - 2.f * dotz[v];
    const float kz = var * __expf(-0.5f * d2 / ls2) + ((m == n) ? JITTER : 0.f);
    Kzz[m * M_IND + n] = kz;
    Dm[m * M_IND + n]  = accS[v] - kz;
  }
}

// ---------------------------------------------------------------------------
// Kernel 2: per 16 X-rows: d2(16x512) via WMMA into LDS, then per-row top-16.
// Both half-waves scan 256 candidates each for their row, then merge the two
// sorted top-16 lists with the bitonic lower-half trick via __shfl_xor(16).
// ---------------------------------------------------------------------------
__global__ void __launch_bounds__(32)
dist_topk_kernel(const float* __restrict__ X, const float* __restrict__ Z,
                 const float* __restrict__ zsq,
                 const float* __restrict__ pls, const float* __restrict__ pvar,
                 int* __restrict__ idxbuf, float* __restrict__ kxzbuf) {
  __shared__ float d2s[16 * M_IND];   // 32 KB
  __shared__ float xsq[16];

  const int blk = blockIdx.x;
  const int lane = threadIdx.x;
  const int half = lane >> 4, l16 = lane & 15;

  if (lane < 16) {
    const int xr = blk * 16 + l16;
    float s = 0.f;
    #pragma unroll
    for (int d = 0; d < DIM; ++d) s = fmaf(X[xr * DIM + d], X[xr * DIM + d], s);
    xsq[l16] = s;
  }

  const int xrow = blk * 16 + l16;
  const v2f a0 = ld2(X + xrow * DIM + half * 2);
  const v2f a1 = ld2(X + xrow * DIM + 4 + half * 2);

  __syncthreads();

  for (int j = 0; j < M_IND / 16; ++j) {
    const int zr = j * 16 + l16;
    v8f dot = {};
    dot = wmma_f32_k4(a0, ld2(Z + zr * DIM + half * 2), dot);
    dot = wmma_f32_k4(a1, ld2(Z + zr * DIM + 4 + half * 2), dot);
    #pragma unroll
    for (int v = 0; v < 8; ++v) {
      const int m = v + half * 8;
      const int n = j * 16 + l16;
      d2s[m * M_IND + n] = xsq[m] + zsq[n] - 2.f * dot[v];
    }
  }

  __syncthreads();

  // --- top-16: 2 lanes per row, 256 candidates each, sorted insertion ---
  const int r = l16;
  const int base = half * (M_IND / 2);
  float bd[K_NN]; int bi[K_NN];
  #pragma unroll
  for (int k = 0; k < K_NN; ++k) { bd[k] = 3.0e38f; bi[k] = 0; }
  for (int c = 0; c < M_IND / 2; ++c) {
    const float d = d2s[r * M_IND + base + c];
    if (d < bd[K_NN - 1]) {
      bd[K_NN - 1] = d; bi[K_NN - 1] = base + c;
      #pragma unroll
      for (int k = K_NN - 1; k >= 1; --k) {
        if (bd[k] < bd[k - 1]) {
          float td = bd[k]; bd[k] = bd[k - 1]; bd[k - 1] = td;
          int   tc = bi[k]; bi[k] = bi[k - 1]; bi[k - 1] = tc;
        }
      }
    }
  }
  // merge the two sorted ascending 16-lists: 16 smallest = min(A[i], B[15-i])
  float md[K_NN]; int mi[K_NN];
  #pragma unroll
  for (int i = 0; i < K_NN; ++i) {
    const float pd = __shfl_xor(bd[K_NN - 1 - i], 16, 32);
    const int   pi = __shfl_xor(bi[K_NN - 1 - i], 16, 32);
    const bool take = (bd[i] <= pd);
    md[i] = take ? bd[i] : pd;
    mi[i] = take ? bi[i] : pi;
  }

  if (lane < 16) {
    const int n = blk * 16 + lane;
    const float ls2 = pls[0] * pls[0];
    const float var = pvar[0];
    #pragma unroll
    for (int k = 0; k < K_NN; ++k) {
      idxbuf[n * K_NN + k] = mi[k];
      kxzbuf[n * K_NN + k] = var * __expf(-0.5f * md[k] / ls2);
    }
  }
}

// ---------------------------------------------------------------------------
// Kernel 3: one wave per point. Gather C = Kzz[idx,idx]+jitter*I and
// A = (S-Kzz)[idx,idx]; in-wave Gauss-Jordan solve C w = kxz (lane = row,
// pivot row broadcast via wave32 shuffles); mean = w.mu_sel,
// cov = var + w^T A w; std = sqrt(max(cov, 0.05)).
// ---------------------------------------------------------------------------
__global__ void __launch_bounds__(256)
solve_kernel(const float* __restrict__ mu, const float* __restrict__ pvar,
             const float* __restrict__ Kzz, const float* __restrict__ Dm,
             const int* __restrict__ idxbuf, const float* __restrict__ kxzbuf,
             float* __restrict__ out) {
  __shared__ float sA[8][256];
  __shared__ float sC[8][256];
  __shared__ float sw[8][16];

  const int wave = threadIdx.x >> 5;
  const int lane = threadIdx.x & 31;
  const int n = blockIdx.x * 8 + wave;
  const int* __restrict__ id = idxbuf + n * K_NN;

  #pragma unroll
  for (int t = 0; t < 8; ++t) {          // 256 elements / 32 lanes
    const int e = lane + t * 32;
    const int r = e >> 4, c = e & 15;
    const int ir = id[r], ic = id[c];    // L2-resident gathers (Kzz/Dm = 2 MB)
    const float kz = Kzz[ir * M_IND + ic];
    sC[wave][e] = kz + ((r == c) ? JITTER : 0.f);
    sA[wave][e] = Dm[ir * M_IND + ic];
  }
  __syncthreads();

  const int r = lane & 15;
  float row[K_NN + 1];
  #pragma unroll
  for (int j = 0; j < K_NN; ++j) row[j] = sC[wave][r * 16 + j];
  row[K_NN] = kxzbuf[n * K_NN + r];

  #pragma unroll
  for (int k = 0; k < K_NN; ++k) {
    float pv[K_NN + 1];
    #pragma unroll
    for (int j = 0; j < K_NN + 1; ++j) pv[j] = __shfl(row[j], k, 32);
    const float f = row[k] / pv[k];
    const bool self = (lane == k);
    #pragma unroll
    for (int j = 0; j < K_NN + 1; ++j) {
      const float nv = row[j] - f * pv[j];
      row[j] = self ? row[j] : nv;
    }
  }
  const float w = row[K_NN] / row[r];
  if (lane < 16) sw[wave][lane] = w;
  __syncthreads();

  float covp = 0.f, meanp = 0.f;
  if (lane < 16) {
    float t = 0.f;
    #pragma unroll
    for (int j = 0; j < K_NN; ++j) t = fmaf(sA[wave][lane * 16 + j], sw[wave][j], t);
    covp  = sw[wave][lane] * t;
    meanp = sw[wave][lane] * mu[id[lane]];
  }
  #pragma unroll
  for (int off = 16; off >= 1; off >>= 1) {
    covp  += __shfl_xor(covp,  off, 32);
    meanp += __shfl_xor(meanp, off, 32);
  }
  if (lane == 0) {
    out[n] = meanp;
    out[N_PTS + n] = sqrtf(fmaxf(pvar[0] + covp, 0.05f));
  }
}

// ---------------------------------------------------------------------------
extern "C" void kernel_launch(void* const* d_in, const int* in_sizes, int n_in,
                              void* d_out, int out_size, void* d_ws, size_t ws_size,
                              hipStream_t stream) {
  const float* X   = (const float*)d_in[0];
  const float* Z   = (const float*)d_in[1];
  const float* Lu  = (const float*)d_in[2];
  const float* mu  = (const float*)d_in[3];
  const float* ls  = (const float*)d_in[4];
  const float* var = (const float*)d_in[5];

  float* ws     = (float*)d_ws;
  float* Kzz    = ws;                               // 512*512 f32 (1 MB)
  float* Dm     = Kzz + M_IND * M_IND;              // 512*512 f32 (1 MB)
  float* Lut    = Dm + M_IND * M_IND;               // 512*512 f32 (1 MB)
  float* zsq    = Lut + M_IND * M_IND;              // 512 f32
  int*   idxbuf = (int*)(zsq + M_IND);              // N*16 i32  (8 MB)
  float* kxzbuf = (float*)(idxbuf + N_PTS * K_NN);  // N*16 f32  (8 MB)
  float* out    = (float*)d_out;                    // [2, N]

  prep_kernel<<<(M_IND * M_IND) / 256, 256, 0, stream>>>(Lu, Z, Lut, zsq);
  dim3 g1(M_IND / 16, M_IND / 16);
  build_mats_kernel<<<g1, 32, 0, stream>>>(Z, Lut, zsq, ls, var, Kzz, Dm);
  dist_topk_kernel<<<N_PTS / 16, 32, 0, stream>>>(X, Z, zsq, ls, var, idxbuf, kxzbuf);
  solve_kernel<<<N_PTS / 8, 256, 0, stream>>>(mu, var, Kzz, Dm, idxbuf, kxzbuf, out);
}